// EdgeGINEConv_39599598469666
// MI455X (gfx1250) — compile-verified
//
#include <hip/hip_runtime.h>

typedef __attribute__((ext_vector_type(2))) float v2f;
typedef __attribute__((ext_vector_type(8))) float v8f;

#define DFEAT 64

// ---------------------------------------------------------------------------
// Phase 1: per-edge message + scatter-add.  msg = relu(x[src] + e), h[dst] += msg
// One thread per (edge, feature).  64 consecutive features of one edge map to
// 64 consecutive lanes -> fully coalesced loads of x-row and edge_attr-row.
// h accumulators (25.6 MB) stay L2-resident (192 MB L2); HW fp32 atomics.
// ---------------------------------------------------------------------------
__global__ __launch_bounds__(256) void gine_edge_kernel(
    const float* __restrict__ x, const int* __restrict__ ei,
    const float* __restrict__ ea, float* __restrict__ h, int E)
{
    long long gid = (long long)blockIdx.x * 256 + threadIdx.x;
    long long total = (long long)E * DFEAT;
    if (gid >= total) return;
    int e = (int)(gid >> 6);
    int d = (int)(gid & 63);
    int src = ei[e];
    int dst = ei[E + e];
    float m = x[(long long)src * DFEAT + d] + ea[(long long)e * DFEAT + d];
    m = fmaxf(m, 0.0f);
    unsafeAtomicAdd(&h[(long long)dst * DFEAT + d], m);   // global_atomic_add_f32
}

// ---------------------------------------------------------------------------
// Phase 2: fused MLP  out = relu(h@W1 + b1)@W2 + b2  using V_WMMA_F32_16X16X4_F32.
// 4 waves / block, each wave owns a 16-row tile.  K=64 -> 16 k-steps of K=4,
// 4 column tiles of N=16.
//
// f32 16x16x4 operand layouts (ISA 7.12.2):
//   A (16x4): lane<16 -> M=lane, v0=K0,v1=K1 ; lane>=16 -> M=lane-16, v0=K2,v1=K3
//   B (4x16): mirrored (v0/v1 = K-pair of lane group), N=lane%16
//   C (16x16): vgpr r, lanes0-15 -> M=r ; lanes16-31 -> M=8+r ; N=lane%16
//
// Weights are pre-swizzled in LDS into exact B-operand order so each inner-loop
// fetch is one contiguous, bank-conflict-free ds_load_b64 straight into the
// even-aligned VGPR pair WMMA needs (kills the v_dual_mov repacking chains).
// Swizzled index: ((kb*4 + nt)*32 + lane) * 2 floats, holding
//   { W[k0][n], W[k0+1][n] }  with  k0 = kb*4 + (lane>>4)*2,  n = nt*16 + lane%16.
// ---------------------------------------------------------------------------
__global__ __launch_bounds__(128) void gine_mlp_kernel(
    const float* __restrict__ h,
    const float* __restrict__ W1, const float* __restrict__ b1,
    const float* __restrict__ W2, const float* __restrict__ b2,
    float* __restrict__ out, int nrows)
{
    __shared__ float sW1[DFEAT * DFEAT];   // swizzled B-operand order
    __shared__ float sW2[DFEAT * DFEAT];   // swizzled B-operand order
    __shared__ float sB1[DFEAT];
    __shared__ float sB2[DFEAT];
    __shared__ float sT[4][16 * DFEAT];    // per-wave staging of the hidden tile

    const int tid  = threadIdx.x;
    const int wave = tid >> 5;
    const int lane = tid & 31;

    // stage weights in swizzled B-operand layout: 2048 pair-entries per matrix
    for (int idx = tid; idx < 16 * 4 * 32; idx += 128) {
        const int l  = idx & 31;
        const int nt = (idx >> 5) & 3;
        const int kb = idx >> 7;
        const int k0 = kb * 4 + (l >> 4) * 2;
        const int n  = nt * 16 + (l & 15);
        sW1[idx * 2 + 0] = W1[(k0 + 0) * DFEAT + n];
        sW1[idx * 2 + 1] = W1[(k0 + 1) * DFEAT + n];
        sW2[idx * 2 + 0] = W2[(k0 + 0) * DFEAT + n];
        sW2[idx * 2 + 1] = W2[(k0 + 1) * DFEAT + n];
    }
    if (tid < DFEAT) { sB1[tid] = b1[tid]; sB2[tid] = b2[tid]; }
    __syncthreads();

    const int row0 = (blockIdx.x * 4 + wave) * 16;
    // keep all waves alive (uniform barriers); clamp addresses, predicate stores
    const int rowb = (row0 + 16 <= nrows) ? row0 : 0;

    const int mrow = lane & 15;   // A-matrix row within tile
    const int kgrp = lane >> 4;   // 0: K pair {0,1}; 1: K pair {2,3} of each k-step
    const int ncol = lane & 15;   // B/C/D column within n-tile

    // per-lane base into swizzled weight arrays (advance by 4*32*2 floats per kb)
    const int wofs = lane * 2;

    // ---- layer 1: A from global h (coalesced b64 loads) ----
    v2f a[16];
    const float* hp = h + (long long)(rowb + mrow) * DFEAT + kgrp * 2;
#pragma unroll
    for (int kb = 0; kb < 16; ++kb)
        a[kb] = *(const v2f*)(hp + kb * 4);

    v8f c0 = {}, c1 = {}, c2 = {}, c3 = {};
#pragma unroll
    for (int kb = 0; kb < 16; ++kb) {
        const float* wb = &sW1[kb * 256 + wofs];
        v2f bb0 = *(const v2f*)(wb +   0);
        v2f bb1 = *(const v2f*)(wb +  64);
        v2f bb2 = *(const v2f*)(wb + 128);
        v2f bb3 = *(const v2f*)(wb + 192);
        c0 = __builtin_amdgcn_wmma_f32_16x16x4_f32(false, a[kb], false, bb0, (short)0, c0, false, false);
        c1 = __builtin_amdgcn_wmma_f32_16x16x4_f32(false, a[kb], false, bb1, (short)0, c1, false, false);
        c2 = __builtin_amdgcn_wmma_f32_16x16x4_f32(false, a[kb], false, bb2, (short)0, c2, false, false);
        c3 = __builtin_amdgcn_wmma_f32_16x16x4_f32(false, a[kb], false, bb3, (short)0, c3, false, false);
    }

    // bias + relu, stage hidden tile to this wave's LDS region
    float* tb = &sT[wave][0];
#pragma unroll
    for (int r = 0; r < 8; ++r) {
        const int M = kgrp * 8 + r;
        tb[M * DFEAT +  0 + ncol] = fmaxf(c0[r] + sB1[ 0 + ncol], 0.0f);
        tb[M * DFEAT + 16 + ncol] = fmaxf(c1[r] + sB1[16 + ncol], 0.0f);
        tb[M * DFEAT + 32 + ncol] = fmaxf(c2[r] + sB1[32 + ncol], 0.0f);
        tb[M * DFEAT + 48 + ncol] = fmaxf(c3[r] + sB1[48 + ncol], 0.0f);
    }
    __syncthreads();

    // ---- layer 2: A from staged LDS tile (contiguous b64 loads) ----
    v2f a2[16];
#pragma unroll
    for (int kb = 0; kb < 16; ++kb)
        a2[kb] = *(const v2f*)(&tb[mrow * DFEAT + kb * 4 + kgrp * 2]);

    v8f d0 = {}, d1 = {}, d2 = {}, d3 = {};
#pragma unroll
    for (int kb = 0; kb < 16; ++kb) {
        const float* wb = &sW2[kb * 256 + wofs];
        v2f bb0 = *(const v2f*)(wb +   0);
        v2f bb1 = *(const v2f*)(wb +  64);
        v2f bb2 = *(const v2f*)(wb + 128);
        v2f bb3 = *(const v2f*)(wb + 192);
        d0 = __builtin_amdgcn_wmma_f32_16x16x4_f32(false, a2[kb], false, bb0, (short)0, d0, false, false);
        d1 = __builtin_amdgcn_wmma_f32_16x16x4_f32(false, a2[kb], false, bb1, (short)0, d1, false, false);
        d2 = __builtin_amdgcn_wmma_f32_16x16x4_f32(false, a2[kb], false, bb2, (short)0, d2, false, false);
        d3 = __builtin_amdgcn_wmma_f32_16x16x4_f32(false, a2[kb], false, bb3, (short)0, d3, false, false);
    }

    if (row0 + 16 <= nrows) {
        float* op = out + (long long)row0 * DFEAT;
#pragma unroll
        for (int r = 0; r < 8; ++r) {
            const int M = kgrp * 8 + r;
            op[M * DFEAT +  0 + ncol] = d0[r] + sB2[ 0 + ncol];
            op[M * DFEAT + 16 + ncol] = d1[r] + sB2[16 + ncol];
            op[M * DFEAT + 32 + ncol] = d2[r] + sB2[32 + ncol];
            op[M * DFEAT + 48 + ncol] = d3[r] + sB2[48 + ncol];
        }
    }
}

// ---------------------------------------------------------------------------
// launch: h(ws) = x ; edge scatter-add into h ; fused WMMA MLP -> out
// ---------------------------------------------------------------------------
extern "C" void kernel_launch(void* const* d_in, const int* in_sizes, int n_in,
                              void* d_out, int out_size, void* d_ws, size_t ws_size,
                              hipStream_t stream) {
    const float* x  = (const float*)d_in[0];
    const int*   ei = (const int*)d_in[1];
    const float* ea = (const float*)d_in[2];
    const float* W1 = (const float*)d_in[3];
    const float* b1 = (const float*)d_in[4];
    const float* W2 = (const float*)d_in[5];
    const float* b2 = (const float*)d_in[6];
    float* out = (float*)d_out;

    const int N = in_sizes[0] / DFEAT;
    const int E = in_sizes[1] / 2;

    float* h = (float*)d_ws;   // [N, 64] accumulator, re-seeded every call

    // h = (1+eps)*x with eps=0  ->  h = x
    hipMemcpyAsync(h, (const void*)x, (size_t)N * DFEAT * sizeof(float),
                   hipMemcpyDeviceToDevice, stream);

    const long long ethreads = (long long)E * DFEAT;
    const int eblocks = (int)((ethreads + 255) / 256);
    gine_edge_kernel<<<eblocks, 256, 0, stream>>>(x, ei, ea, h, E);

    const int tiles   = (N + 15) / 16;
    const int mblocks = (tiles + 3) / 4;
    gine_mlp_kernel<<<mblocks, 128, 0, stream>>>(h, W1, b1, W2, b2, out, N);
}